// Warper_57896159150706
// MI455X (gfx1250) — compile-verified
//
#include <hip/hip_runtime.h>
#include <math.h>

typedef float v2f __attribute__((ext_vector_type(2)));
typedef float v8f __attribute__((ext_vector_type(8)));

#define IMG_H 512
#define IMG_W 512
#define NB 32
#define NC 3
#define PLANE (IMG_H * IMG_W)

// ---------------------------------------------------------------------------
// Kernel 1: per-batch affine params (closed form of inv(fw) @ inv(S) @ fw,
// pre-composed with normalized-grid <-> pixel coordinate mappings).
// params[b] = {ax, bx, cx, ay, by, cy, 0, 0}:  sx = ax*w + bx*h + cx, etc.
// ---------------------------------------------------------------------------
__global__ void prep_params(const float* __restrict__ thetas,
                            const float* __restrict__ l1s,
                            const float* __restrict__ l2s,
                            float* __restrict__ params) {
  int b = threadIdx.x;
  if (b >= NB) return;
  float th = thetas[b];
  float c = cosf(th);        // cos(-th) == cos(th)
  float s = -sinf(th);       // sin(-th)
  float i1 = 1.0f / l1s[b];
  float i2 = 1.0f / l2s[b];
  // M = R^T * diag(i1,i2) * R  (symmetric)
  float m00 = c * c * i1 + s * s * i2;
  float m01 = c * s * (i2 - i1);
  float m10 = m01;
  float m11 = s * s * i1 + c * c * i2;
  const float Wf = (float)IMG_W, Hf = (float)IMG_H;
  // translation v = (M - I) [W; H]
  float v0 = (m00 - 1.0f) * Wf + m01 * Hf;
  float v1 = m10 * Wf + (m11 - 1.0f) * Hf;
  // Fold normalized-grid math into pixel-space affine:
  //   X(w) = (w+0.5)*(2/W)-1 ; gx = m00*X + m01*Y + v0 ; sx = (gx+1)*W/2 - 0.5
  float ax = m00;
  float bx = m01 * (Wf / Hf);
  float gx0 = m00 * (1.0f / Wf - 1.0f) + m01 * (1.0f / Hf - 1.0f) + v0;
  float cx = (gx0 + 1.0f) * (Wf * 0.5f) - 0.5f;
  float ay = m10 * (Hf / Wf);
  float by = m11;
  float gy0 = m10 * (1.0f / Wf - 1.0f) + m11 * (1.0f / Hf - 1.0f) + v1;
  float cy = (gy0 + 1.0f) * (Hf * 0.5f) - 0.5f;
  float* p = params + b * 8;
  p[0] = ax; p[1] = bx; p[2] = cx;
  p[3] = ay; p[4] = by; p[5] = cy;
  p[6] = 0.0f; p[7] = 0.0f;
}

// ---------------------------------------------------------------------------
// Kernel 2: one wave32 owns one 16x16 output tile for one batch (all 3 chans).
// Grid coordinates for the whole tile are produced by two
// V_WMMA_F32_16X16X4_F32 ops:  D[h][w] = (b*h + c)*1 + 1*(a*w).
// WMMA D layout: VGPR r, lane l -> h = r + 8*(l>=16), w = l%16, so each lane
// directly owns 8 pixels of the tile with no cross-lane shuffles.
// ---------------------------------------------------------------------------
__global__ __launch_bounds__(256) void warp_sample(
    const float* __restrict__ x,
    const float* __restrict__ params,
    float* __restrict__ out) {
  const int lane = threadIdx.x & 31;
  const int wave = threadIdx.x >> 5;
  const int tile = blockIdx.x * 8 + wave;      // 32768 tiles total
  const int b    = tile >> 10;                 // tile / 1024
  const int rem  = tile & 1023;
  const int tile_h = (rem >> 5) << 4;
  const int tile_w = (rem & 31) << 4;

  const float* pp = params + b * 8;
  const float ax = pp[0], bx = pp[1], cx = pp[2];
  const float ay = pp[3], by = pp[4], cy = pp[5];

  const int  n  = lane & 15;
  const bool lo = lane < 16;
  const float nf = (float)n;

  // A (16x4 f32): lane m%16 holds row m; VGPR0 = K0 (lanes 0-15) | K2 (16-31),
  // VGPR1 = K1 | K3.  Row h = [b*h + c, 1, 0, 0].
  v2f a_x, a_y;
  a_x.x = lo ? (bx * ((float)tile_h + nf) + cx) : 0.0f;
  a_x.y = lo ? 1.0f : 0.0f;
  a_y.x = lo ? (by * ((float)tile_h + nf) + cy) : 0.0f;
  a_y.y = a_x.y;

  // B (4x16 f32): VGPR0 = row K0 (lanes 0-15) | row K2 (16-31); VGPR1 = K1|K3.
  // Col w = [1, a*w, 0, 0].
  v2f b_x, b_y;
  b_x.x = lo ? 1.0f : 0.0f;
  b_x.y = lo ? (ax * ((float)tile_w + nf)) : 0.0f;
  b_y.x = b_x.x;
  b_y.y = lo ? (ay * ((float)tile_w + nf)) : 0.0f;

  v8f zero = {};
  v8f dx = __builtin_amdgcn_wmma_f32_16x16x4_f32(
      false, a_x, false, b_x, (short)0, zero, false, false);
  v8f dy = __builtin_amdgcn_wmma_f32_16x16x4_f32(
      false, a_y, false, b_y, (short)0, zero, false, false);

  const int w      = tile_w + n;
  const int h_base = tile_h + (lo ? 0 : 8);
  const float* xb = x   + (size_t)b * (NC * PLANE);
  float*       ob = out + (size_t)b * (NC * PLANE);

#pragma unroll
  for (int r = 0; r < 8; ++r) {
    float sx = dx[r];
    float sy = dy[r];
    float x0f = floorf(sx), y0f = floorf(sy);
    float wx1 = sx - x0f, wx0 = 1.0f - wx1;
    float wy1 = sy - y0f, wy0 = 1.0f - wy1;
    int x0 = (int)x0f, y0 = (int)y0f;
    int x1 = x0 + 1,  y1 = y0 + 1;
    // zero-padding masks folded into the bilinear weights
    float mx0 = (x0 >= 0 && x0 < IMG_W) ? 1.0f : 0.0f;
    float mx1 = (x1 >= 0 && x1 < IMG_W) ? 1.0f : 0.0f;
    float my0 = (y0 >= 0 && y0 < IMG_H) ? 1.0f : 0.0f;
    float my1 = (y1 >= 0 && y1 < IMG_H) ? 1.0f : 0.0f;
    int x0c = min(max(x0, 0), IMG_W - 1);
    int x1c = min(max(x1, 0), IMG_W - 1);
    int y0c = min(max(y0, 0), IMG_H - 1);
    int y1c = min(max(y1, 0), IMG_H - 1);
    int i00 = y0c * IMG_W + x0c;
    int i01 = y0c * IMG_W + x1c;
    int i10 = y1c * IMG_W + x0c;
    int i11 = y1c * IMG_W + x1c;
    float w00 = wy0 * wx0 * my0 * mx0;
    float w01 = wy0 * wx1 * my0 * mx1;
    float w10 = wy1 * wx0 * my1 * mx0;
    float w11 = wy1 * wx1 * my1 * mx1;
    int oidx = (h_base + r) * IMG_W + w;
#pragma unroll
    for (int c = 0; c < NC; ++c) {
      const float* p = xb + c * PLANE;
      float v = w00 * p[i00] + w01 * p[i01] + w10 * p[i10] + w11 * p[i11];
      ob[c * PLANE + oidx] = v;
    }
  }
}

// ---------------------------------------------------------------------------
extern "C" void kernel_launch(void* const* d_in, const int* in_sizes, int n_in,
                              void* d_out, int out_size, void* d_ws,
                              size_t ws_size, hipStream_t stream) {
  (void)in_sizes; (void)n_in; (void)out_size; (void)ws_size;
  const float* x      = (const float*)d_in[0];
  const float* thetas = (const float*)d_in[1];
  const float* l1s    = (const float*)d_in[2];
  const float* l2s    = (const float*)d_in[3];
  float* out    = (float*)d_out;
  float* params = (float*)d_ws;   // NB * 8 floats = 1 KB

  prep_params<<<1, 32, 0, stream>>>(thetas, l1s, l2s, params);

  // tiles = NB * (512/16) * (512/16) = 32768; 8 waves (tiles) per 256-thr block
  const int blocks = (NB * (IMG_H / 16) * (IMG_W / 16)) / 8;
  warp_sample<<<blocks, 256, 0, stream>>>(x, params, out);
}